// IterativeRecoModel_81209241633555
// MI455X (gfx1250) — compile-verified
//
#include <hip/hip_runtime.h>
#include <math.h>

#define HH 512
#define WW 512
#define N_ANGLES 360
#define N_DET 736
#define N_STEPS 725
#define SINO_SIZE (N_ANGLES * N_DET)   /* 264960 */
#define IMG_SIZE (HH * WW)             /* 262144 */

typedef __attribute__((ext_vector_type(2))) float v2f;
typedef __attribute__((ext_vector_type(8))) float v8f;

// ---------------------------------------------------------------------------
// Kernel 1: updated_reco = x + reco   (float4-vectorized, 262144 elements)
// ---------------------------------------------------------------------------
__global__ __launch_bounds__(256) void reco_add_kernel(
    const float4* __restrict__ x, const float4* __restrict__ reco,
    float4* __restrict__ out_reco) {
  const int i = blockIdx.x * 256 + threadIdx.x;   // 256 blocks * 256 thr * 4 = 262144
  const float4 a = x[i];
  const float4 b = reco[i];
  out_reco[i] = make_float4(a.x + b.x, a.y + b.y, a.z + b.z, a.w + b.w);
}

// ---------------------------------------------------------------------------
// Kernel 2: ray-driven Radon forward projection.
//   grid = (23 detector tiles of 32, 360 angles), block = 256 (8 wave32)
//   lane  -> detector within tile
//   wave  -> 1/8th of the 725 ray steps
//   cross-wave reduction done with V_WMMA_F32_16X16X4_F32 against ones (exact)
// ---------------------------------------------------------------------------
__global__ __launch_bounds__(256) void radon_kernel(
    const float* __restrict__ img, const float* __restrict__ angles,
    float* __restrict__ sino) {
  __shared__ float part[8][32];

  const int tid   = threadIdx.x;
  const int lane  = tid & 31;
  const int wave  = tid >> 5;
  const int dtile = blockIdx.x;          // 0..22
  const int ang   = blockIdx.y;          // 0..359
  const int det   = dtile * 32 + lane;

  const float theta = angles[ang];
  const float c = cosf(theta);
  const float s = sinf(theta);

  const float u  = ((float)det - (N_DET - 1) * 0.5f);  // DET_SPACING = 1
  const float cx = (WW - 1) * 0.5f;
  const float cy = (HH - 1) * 0.5f;

  // Per-wave contiguous t-chunk
  const int chunk = (N_STEPS + 7) / 8;                 // 91
  const int k0 = wave * chunk;
  const int k1 = (k0 + chunk < N_STEPS) ? (k0 + chunk) : N_STEPS;

  const float tbase = (float)k0 - (N_STEPS - 1) * 0.5f; // STEP = 1
  const float xs0 = fmaf(c, tbase, fmaf(-s, u, cx));
  const float ys0 = fmaf(s, tbase, fmaf(c, u, cy));

  // CDNA5 global_prefetch_b8 — warm L2/L0 with a strided slice of the image
  __builtin_prefetch(img + ((tid * 977) & (IMG_SIZE - 1)), 0, 3);

  float acc = 0.0f;
  for (int k = k0; k < k1; ++k) {
    const float kf = (float)(k - k0);
    const float xs = fmaf(c, kf, xs0);
    const float ys = fmaf(s, kf, ys0);

    const float fx = floorf(xs);
    const float fy = floorf(ys);
    const float wx = xs - fx;
    const float wy = ys - fy;
    const int ix = (int)fx;
    const int iy = (int)fy;
    const int ix1 = ix + 1;
    const int iy1 = iy + 1;

    // clamped fetch indices (v_med3-style)
    const int cx0 = min(max(ix, 0), WW - 1);
    const int cx1 = min(max(ix1, 0), WW - 1);
    const int cy0 = min(max(iy, 0), HH - 1);
    const int cy1 = min(max(iy1, 0), HH - 1);

    const float v00r = img[cy0 * WW + cx0];
    const float v01r = img[cy0 * WW + cx1];
    const float v10r = img[cy1 * WW + cx0];
    const float v11r = img[cy1 * WW + cx1];

    // zero-padding masks via unsigned compare (covers <0 and >=dim)
    const bool bx0 = (unsigned)ix  < (unsigned)WW;
    const bool bx1 = (unsigned)ix1 < (unsigned)WW;
    const bool by0 = (unsigned)iy  < (unsigned)HH;
    const bool by1 = (unsigned)iy1 < (unsigned)HH;

    const float v00 = (bx0 && by0) ? v00r : 0.0f;
    const float v01 = (bx1 && by0) ? v01r : 0.0f;
    const float v10 = (bx0 && by1) ? v10r : 0.0f;
    const float v11 = (bx1 && by1) ? v11r : 0.0f;

    const float r0 = fmaf(wx, v01 - v00, v00);
    const float r1 = fmaf(wx, v11 - v10, v10);
    acc += fmaf(wy, r1 - r0, r0);
  }

  part[wave][lane] = acc;
  __syncthreads();

  // -------------------------------------------------------------------------
  // WMMA reduction: rows = 16 detectors, K = 4 wave-partials per instruction.
  // B = all-ones -> D[m,n] = sum_k A[m,k] + C[m,n]; chain C to cover 8 waves.
  // Every wave executes with full EXEC (WMMA requirement); wave 0 stores.
  // A 16x4 f32 layout: lanes 0-15 hold K={0,1}, lanes 16-31 hold K={2,3}.
  // -------------------------------------------------------------------------
  const int row = lane & 15;
  const int kb  = (lane >> 4) * 2;      // 0 or 2
  v2f ones;
  ones.x = 1.0f;
  ones.y = 1.0f;

  float* __restrict__ out_row = sino + ang * N_DET + dtile * 32;

#pragma unroll
  for (int doff = 0; doff < 32; doff += 16) {
    v2f a1, a2;
    a1.x = part[kb + 0][doff + row];    // waves 0..3 as K-slices
    a1.y = part[kb + 1][doff + row];
    a2.x = part[kb + 4][doff + row];    // waves 4..7 as K-slices
    a2.y = part[kb + 5][doff + row];

    v8f d = {};
    d = __builtin_amdgcn_wmma_f32_16x16x4_f32(false, a1, false, ones,
                                              (short)0, d, false, false);
    d = __builtin_amdgcn_wmma_f32_16x16x4_f32(false, a2, false, ones,
                                              (short)0, d, false, false);

    // D columns are identical; lane 0 holds rows 0-7, lane 16 holds rows 8-15
    if (wave == 0 && (lane & 15) == 0) {
      const int mbase = (lane >> 4) * 8;
#pragma unroll
      for (int m = 0; m < 8; ++m) {
        out_row[doff + mbase + m] = d[m];   // * STEP (== 1.0)
      }
    }
  }
}

// ---------------------------------------------------------------------------
// Launch
// ---------------------------------------------------------------------------
extern "C" void kernel_launch(void* const* d_in, const int* in_sizes, int n_in,
                              void* d_out, int out_size, void* d_ws,
                              size_t ws_size, hipStream_t stream) {
  (void)in_sizes; (void)n_in; (void)out_size; (void)d_ws; (void)ws_size;

  const float* x      = (const float*)d_in[0];
  const float* reco   = (const float*)d_in[1];
  const float* angles = (const float*)d_in[2];

  float* sino     = (float*)d_out;              // 360*736
  float* out_reco = (float*)d_out + SINO_SIZE;  // 512*512

  // updated_reco = x + reco (also makes the image L2-resident for the radon pass)
  reco_add_kernel<<<IMG_SIZE / (256 * 4), 256, 0, stream>>>(
      (const float4*)x, (const float4*)reco, (float4*)out_reco);

  dim3 grid(N_DET / 32, N_ANGLES);  // 23 x 360
  radon_kernel<<<grid, 256, 0, stream>>>(out_reco, angles, sino);
}